// AdaptiveMixGNNLayer_17987323036319
// MI455X (gfx1250) — compile-verified
//
#include <hip/hip_runtime.h>
#include <hip/hip_bf16.h>

typedef float v2f __attribute__((ext_vector_type(2)));
typedef float v8f __attribute__((ext_vector_type(8)));

#define GNN_N_NODES 50000
#define GNN_N_EDGES 640000
#define GNN_D       128
#define EDGES_PER_WAVE 32

// ---------------------------------------------------------------------------
// Kernel 1: zero the two [N,128] f32 accumulators living in d_ws.
// Must run every launch (we accumulate with atomics -> need clean slate).
// ---------------------------------------------------------------------------
__global__ void gnn_zero_ws(float* __restrict__ ws, int n) {
  int i = blockIdx.x * blockDim.x + threadIdx.x;
  int stride = gridDim.x * blockDim.x;
  for (; i < n; i += stride) ws[i] = 0.0f;
}

// ---------------------------------------------------------------------------
// Kernel 2/3: SpMM  acc[rows[e], :] += vals[e] * x[cols[e], :]
// rows is SORTED -> each wave walks a contiguous strip of edges, keeps the
// running row-sum in registers (float4 per lane, 32 lanes * 4 = D=128) and
// only flushes with global_atomic_add_f32 when the destination row changes.
// x (25.6MB) is L2-resident on MI455X (192MB L2); prefetch next gather row.
// ---------------------------------------------------------------------------
__global__ void gnn_spmm(const int*   __restrict__ rows,
                         const int*   __restrict__ cols,
                         const float* __restrict__ vals,
                         const float* __restrict__ x,
                         float*       __restrict__ acc_out,
                         int n_edges) {
  const int lane = threadIdx.x & 31;
  const int wave = (int)((blockIdx.x * blockDim.x + threadIdx.x) >> 5);
  long e0 = (long)wave * EDGES_PER_WAVE;
  if (e0 >= n_edges) return;                 // wave-uniform exit
  long e1 = e0 + EDGES_PER_WAVE;
  if (e1 > n_edges) e1 = n_edges;

  const int d0 = lane * 4;                   // this lane's 4 channels
  float4 acc = make_float4(0.f, 0.f, 0.f, 0.f);
  int cur = rows[e0];

  for (long e = e0; e < e1; ++e) {
    const int   r = rows[e];
    const int   c = cols[e];
    const float v = vals[e];

    if (e + 1 < e1) {                        // hide gather latency
      const int cn = cols[e + 1];
      __builtin_prefetch(&x[(long)cn * GNN_D + d0], 0, 3);
    }

    if (r != cur) {                          // row boundary -> flush
      float* dst = acc_out + (long)cur * GNN_D + d0;
      atomicAdd(dst + 0, acc.x);
      atomicAdd(dst + 1, acc.y);
      atomicAdd(dst + 2, acc.z);
      atomicAdd(dst + 3, acc.w);
      acc = make_float4(0.f, 0.f, 0.f, 0.f);
      cur = r;
    }

    const float4 xr = *(const float4*)(x + (long)c * GNN_D + d0);
    acc.x += v * xr.x;
    acc.y += v * xr.y;
    acc.z += v * xr.z;
    acc.w += v * xr.w;
  }

  float* dst = acc_out + (long)cur * GNN_D + d0;
  atomicAdd(dst + 0, acc.x);
  atomicAdd(dst + 1, acc.y);
  atomicAdd(dst + 2, acc.z);
  atomicAdd(dst + 3, acc.w);
}

// ---------------------------------------------------------------------------
// Kernel 4: fused  out = relu(alpha*(A_lp @ W_lp^T) + (1-alpha)*(A_hp @ W_hp^T) + bias)
// One wave per 16x16 output tile using V_WMMA_F32_16X16X4_F32 (full f32 path,
// matches the f32 reference). K=128 stepped by 4 -> 32 WMMAs per weight matrix.
// A 16x4 f32 layout: lane<16 -> M=lane,   K = k0+{0,1}
//                    lane>=16 -> M=lane-16, K = k0+{2,3}
// B 4x16 f32 layout (symmetric): lane<16 -> N=lane, K = k0+{0,1}; upper half K+2
// B[k][n] = W[n][k] (einsum 'nd,od->no'), so lane n loads W row n at offset k.
// C/D: VGPR r -> M = 16*tile_m + r + (lane>=16 ? 8 : 0), N = 16*tile_n + (lane&15)
// ---------------------------------------------------------------------------
__global__ void gnn_wmma_mix(const float* __restrict__ A_lp,
                             const float* __restrict__ A_hp,
                             const float* __restrict__ W_lp,
                             const float* __restrict__ W_hp,
                             const float* __restrict__ bias,
                             const float* __restrict__ alpha_raw,
                             float*       __restrict__ out) {
  const int lane = threadIdx.x & 31;
  const int wave_in_block = threadIdx.x >> 5;
  const int tile   = blockIdx.x * 8 + wave_in_block;  // 25000 tiles total
  const int tile_m = tile >> 3;                       // 0..3124  (node tiles)
  const int tile_n = tile & 7;                        // 0..7     (channel tiles)
  const int half   = lane >> 4;                       // 0 or 1
  const int l15    = lane & 15;

  const int m_node = tile_m * 16 + l15;               // A row this lane feeds
  const int o_col  = tile_n * 16 + l15;               // B column (out channel)

  const float* a_lp_row = A_lp + (long)m_node * GNN_D;
  const float* a_hp_row = A_hp + (long)m_node * GNN_D;
  const float* w_lp_row = W_lp + (long)o_col  * GNN_D;
  const float* w_hp_row = W_hp + (long)o_col  * GNN_D;

  v8f c_lp = {};
  v8f c_hp = {};

#pragma unroll 4
  for (int k0 = 0; k0 < GNN_D; k0 += 4) {
    const int kk = k0 + half * 2;
    v2f a0 = *(const v2f*)(a_lp_row + kk);
    v2f b0 = *(const v2f*)(w_lp_row + kk);
    v2f a1 = *(const v2f*)(a_hp_row + kk);
    v2f b1 = *(const v2f*)(w_hp_row + kk);
    // 8 args: (neg_a, A, neg_b, B, c_mod, C, reuse_a, reuse_b)
    c_lp = __builtin_amdgcn_wmma_f32_16x16x4_f32(false, a0, false, b0,
                                                 (short)0, c_lp, false, false);
    c_hp = __builtin_amdgcn_wmma_f32_16x16x4_f32(false, a1, false, b1,
                                                 (short)0, c_hp, false, false);
  }

  const float ar    = alpha_raw[0];
  const float alpha = 1.0f / (1.0f + __expf(-ar));
  const float bsc   = bias[o_col];
  const int row_base = tile_m * 16 + half * 8;

#pragma unroll
  for (int r = 0; r < 8; ++r) {
    float v = alpha * c_lp[r] + (1.0f - alpha) * c_hp[r] + bsc;
    v = v > 0.0f ? v : 0.0f;
    out[(long)(row_base + r) * GNN_D + o_col] = v;
  }
}

// ---------------------------------------------------------------------------
// Launch: in_sizes order = x, lp_rows, lp_cols, lp_vals, hp_rows, hp_cols,
//         hp_vals, W_LP, W_HP, bias, alpha_raw.
// d_ws layout: [0, N*D)  = LP accumulator, [N*D, 2*N*D) = HP accumulator
// (needs 2*50000*128*4 = 51.2 MB of workspace).
// ---------------------------------------------------------------------------
extern "C" void kernel_launch(void* const* d_in, const int* in_sizes, int n_in,
                              void* d_out, int out_size, void* d_ws, size_t ws_size,
                              hipStream_t stream) {
  const float* x        = (const float*)d_in[0];
  const int*   lp_rows  = (const int*)  d_in[1];
  const int*   lp_cols  = (const int*)  d_in[2];
  const float* lp_vals  = (const float*)d_in[3];
  const int*   hp_rows  = (const int*)  d_in[4];
  const int*   hp_cols  = (const int*)  d_in[5];
  const float* hp_vals  = (const float*)d_in[6];
  const float* W_lp     = (const float*)d_in[7];
  const float* W_hp     = (const float*)d_in[8];
  const float* bias     = (const float*)d_in[9];
  const float* alpha    = (const float*)d_in[10];
  float*       out      = (float*)d_out;

  float* ws     = (float*)d_ws;
  float* acc_lp = ws;
  float* acc_hp = ws + (long)GNN_N_NODES * GNN_D;

  // 1) zero accumulators (12.8M floats)
  const int n_acc = 2 * GNN_N_NODES * GNN_D;
  gnn_zero_ws<<<2048, 256, 0, stream>>>(ws, n_acc);

  // 2+3) SpMM: 640000 edges / 32 per wave = 20000 waves = 2500 blocks of 8 waves
  const int spmm_blocks = (GNN_N_EDGES / EDGES_PER_WAVE + 7) / 8;  // 2500
  gnn_spmm<<<spmm_blocks, 256, 0, stream>>>(lp_rows, lp_cols, lp_vals, x,
                                            acc_lp, GNN_N_EDGES);
  gnn_spmm<<<spmm_blocks, 256, 0, stream>>>(hp_rows, hp_cols, hp_vals, x,
                                            acc_hp, GNN_N_EDGES);

  // 4) fused WMMA GEMM + sigmoid-mix + bias + relu
  // tiles = (50000/16) * (128/16) = 3125 * 8 = 25000; 8 waves/block -> 3125 blocks
  gnn_wmma_mix<<<3125, 256, 0, stream>>>(acc_lp, acc_hp, W_lp, W_hp,
                                         bias, alpha, out);
}